// QuantumAttention_65481071408308
// MI455X (gfx1250) — compile-verified
//
#include <hip/hip_runtime.h>
#include <hip/hip_bf16.h>

typedef float v2f __attribute__((ext_vector_type(2)));
typedef float v8f __attribute__((ext_vector_type(8)));

#define B_DIM 8
#define L_DIM 2048
#define E_DIM 512
#define Q_DIM 8

// Fast sigmoid: v_exp_f32 + v_rcp_f32 (hardware trans ops), no IEEE div sequence.
__device__ __forceinline__ float fast_sigmoid(float x)
{
    return __builtin_amdgcn_rcpf(1.0f + __expf(-x));
}

// ---------------------------------------------------------------------------
// Kernel 1: rot[b,l,q] = sum_e x[b,l,e] * rotation[e,q]
// One wave per row (16384 rows). rotation transposed into LDS [q][e] so the
// inner loop reads are conflict-free across lanes. Coalesced x reads.
// ---------------------------------------------------------------------------
__global__ __launch_bounds__(256)
void qa_rot_kernel(const float* __restrict__ x,
                   const float* __restrict__ rotation,
                   float* __restrict__ rot)
{
    __shared__ float rl[Q_DIM * E_DIM];   // rl[q*512 + e] = rotation[e*8 + q]

    const int t = threadIdx.x;
    #pragma unroll
    for (int k = 0; k < 16; ++k) {
        int idx = k * 256 + t;            // 0..4095
        int q = idx >> 9;
        int e = idx & 511;
        rl[idx] = rotation[e * Q_DIM + q];
    }
    __syncthreads();

    const int wave = t >> 5;
    const int lane = t & 31;
    const int row  = blockIdx.x * 8 + wave;       // 0..16383
    const float* xr = x + (size_t)row * E_DIM;

    float acc[Q_DIM] = {};
    for (int k = 0; k < 16; ++k) {
        int c = lane + 32 * k;
        float xv = xr[c];                 // coalesced 128B per wave
        #pragma unroll
        for (int q = 0; q < Q_DIM; ++q)
            acc[q] += xv * rl[q * E_DIM + c];
    }

    #pragma unroll
    for (int q = 0; q < Q_DIM; ++q) {
        float s = acc[q];
        s += __shfl_xor(s, 16);
        s += __shfl_xor(s, 8);
        s += __shfl_xor(s, 4);
        s += __shfl_xor(s, 2);
        s += __shfl_xor(s, 1);
        if (lane == q) rot[(size_t)row * Q_DIM + q] = s;
    }
}

// ---------------------------------------------------------------------------
// Kernel 2: per batch b, S = sigmoid(rot_b . rot_b^T), row-normalized.
// One WG = 16-row strip. Each wave computes 16x16 tiles with
// V_WMMA_F32_16X16X4_F32 (two WMMAs accumulate K=8).
// Pass 1: row sums (shfl_xor reduce + LDS float atomics).
// Pass 2: recompute, scale by 1/rowsum, store.
// ---------------------------------------------------------------------------
__global__ __launch_bounds__(256)
void qa_attn_kernel(const float* __restrict__ rot, float* __restrict__ out)
{
    __shared__ float rowsum[16];
    __shared__ float rowinv[16];

    const int t    = threadIdx.x;
    const int wave = t >> 5;
    const int lane = t & 31;
    const int mr   = lane & 15;     // row/col-in-tile for this lane
    const int kh   = lane >> 4;     // K half (0: K=0,1 ; 1: K=2,3)

    const int wg = blockIdx.x;      // 0..1023
    const int b  = wg >> 7;
    const int i0 = (wg & 127) << 4; // strip start row

    const float* rb = rot + (size_t)b * L_DIM * Q_DIM;

    if (t < 16) rowsum[t] = 0.0f;
    __syncthreads();

    // A fragment (shared by all 8 waves): rows i0+mr, cols 2*kh (+4)
    const float* arow = rb + (size_t)(i0 + mr) * Q_DIM + 2 * kh;
    v2f a0 = *(const v2f*)(arow);
    v2f a1 = *(const v2f*)(arow + 4);

    // ---------------- pass 1: row sums ----------------
    float psum[8] = {};
    for (int jt = 0; jt < 16; ++jt) {
        const int j0 = (((jt << 3) + wave) << 4);
        const float* brow = rb + (size_t)(j0 + mr) * Q_DIM + 2 * kh;
        v2f b0 = *(const v2f*)(brow);
        v2f b1 = *(const v2f*)(brow + 4);
        v8f acc = {0.f, 0.f, 0.f, 0.f, 0.f, 0.f, 0.f, 0.f};
        acc = __builtin_amdgcn_wmma_f32_16x16x4_f32(false, a0, false, b0,
                                                    (short)0, acc, false, false);
        acc = __builtin_amdgcn_wmma_f32_16x16x4_f32(false, a1, false, b1,
                                                    (short)0, acc, false, false);
        #pragma unroll
        for (int v = 0; v < 8; ++v)
            psum[v] += fast_sigmoid(acc[v]);
    }
    #pragma unroll
    for (int v = 0; v < 8; ++v) {
        float s = psum[v];
        s += __shfl_xor(s, 8);          // reduce over the 16 mr-lanes (stay in kh half)
        s += __shfl_xor(s, 4);
        s += __shfl_xor(s, 2);
        s += __shfl_xor(s, 1);
        if (mr == v) atomicAdd(&rowsum[(kh << 3) + v], s);
    }
    __syncthreads();
    if (t < 16) rowinv[t] = __builtin_amdgcn_rcpf(rowsum[t]);
    __syncthreads();

    float inv[8];
    #pragma unroll
    for (int v = 0; v < 8; ++v) inv[v] = rowinv[(kh << 3) + v];

    // ---------------- pass 2: normalize + store ----------------
    float* ob = out + (size_t)b * L_DIM * L_DIM + (size_t)i0 * L_DIM;
    for (int jt = 0; jt < 16; ++jt) {
        const int j0 = (((jt << 3) + wave) << 4);
        const float* brow = rb + (size_t)(j0 + mr) * Q_DIM + 2 * kh;
        v2f b0 = *(const v2f*)(brow);
        v2f b1 = *(const v2f*)(brow + 4);
        v8f acc = {0.f, 0.f, 0.f, 0.f, 0.f, 0.f, 0.f, 0.f};
        acc = __builtin_amdgcn_wmma_f32_16x16x4_f32(false, a0, false, b0,
                                                    (short)0, acc, false, false);
        acc = __builtin_amdgcn_wmma_f32_16x16x4_f32(false, a1, false, b1,
                                                    (short)0, acc, false, false);
        #pragma unroll
        for (int v = 0; v < 8; ++v) {
            float s = fast_sigmoid(acc[v]) * inv[v];
            int r = (kh << 3) + v;
            ob[(size_t)r * L_DIM + j0 + mr] = s;   // 64B-contiguous per half-wave
        }
    }
}

// ---------------------------------------------------------------------------
extern "C" void kernel_launch(void* const* d_in, const int* in_sizes, int n_in,
                              void* d_out, int out_size, void* d_ws, size_t ws_size,
                              hipStream_t stream)
{
    const float* x        = (const float*)d_in[0];   // (8, 2048, 512) f32
    const float* rotation = (const float*)d_in[1];   // (512, 8) f32
    // d_in[2] (entangle) unused in forward, matching the reference.
    float* outp = (float*)d_out;                     // (8, 2048, 2048) f32
    float* rot  = (float*)d_ws;                      // (8, 2048, 8) f32 = 512 KB scratch

    // Kernel 1: 16384 rows, 8 rows (waves) per 256-thread block.
    qa_rot_kernel<<<(B_DIM * L_DIM) / 8, 256, 0, stream>>>(x, rotation, rot);

    // Kernel 2: 8 batches x 128 sixteen-row strips.
    qa_attn_kernel<<<B_DIM * (L_DIM / 16), 256, 0, stream>>>(rot, outp);
}